// log_38156489458323
// MI455X (gfx1250) — compile-verified
//
#include <hip/hip_runtime.h>
#include <cstdint>

typedef __attribute__((ext_vector_type(2))) float v2f;
typedef __attribute__((ext_vector_type(8))) float v8f;

#define TDIM 1024
#define DDIM 64
#define NW 4              // waves per workgroup
#define BLOCK (NW * 32)

// One workgroup per (batch b, 16-row q tile i).
// Wave w computes causal score tiles j = w, w+NW, ... (j <= i) with
// v_wmma_f32_16x16x4_f32 chains into LDS; then cooperative in-place softmax;
// then one streaming write of dropout(probs).
__global__ __launch_bounds__(BLOCK) void attn_probs_dropout_kernel(
    const float* __restrict__ q, const float* __restrict__ k,
    float* __restrict__ out)
{
    __shared__ float sc[16][TDIM];   // 64 KB score/prob tile for this row block

    const int i    = blockIdx.x;     // q row tile index (0..T/16-1)
    const int b    = blockIdx.y;     // batch
    const int t    = threadIdx.x;
    const int wave = t >> 5;
    const int lane = t & 31;
    const int half = lane >> 4;      // 0: lanes 0-15, 1: lanes 16-31
    const int l16  = lane & 15;
    const int kb   = half * 2;       // K sub-offset per ISA fp32 A/B layout

    // ---- A fragments (q tile, 16x64 fp32), reused across all column tiles ----
    // A 16x4 layout: lanes 0-15 -> M=lane, VGPR0/1 = K0/K1; lanes 16-31 -> K2/K3.
    const float* qrow = q + ((size_t)(b * TDIM + i * 16 + l16)) * DDIM;
    v2f afrag[16];
    #pragma unroll
    for (int s = 0; s < 16; ++s) {
        afrag[s].x = qrow[4 * s + kb];
        afrag[s].y = qrow[4 * s + kb + 1];
    }

    const float scale = 0.125f;      // D^-1/2 = 1/sqrt(64)

    // ---- causal score tiles: j <= i (wave-uniform loop => EXEC all ones) ----
    for (int j = wave; j <= i; j += NW) {
        // B 4x16 layout (transpose-symmetric striping): lane = N, VGPR pair = K.
        // B[K=d][N=n] = k[j*16+n][d]
        const float* krow = k + ((size_t)(b * TDIM + j * 16 + l16)) * DDIM;
        v8f c = {};
        #pragma unroll
        for (int s = 0; s < 16; ++s) {
            v2f bfrag;
            bfrag.x = krow[4 * s + kb];
            bfrag.y = krow[4 * s + kb + 1];
            c = __builtin_amdgcn_wmma_f32_16x16x4_f32(
                /*neg_a=*/false, afrag[s], /*neg_b=*/false, bfrag,
                /*c_mod=*/(short)0, c, /*reuse_a=*/false, /*reuse_b=*/false);
        }
        // C/D 16x16 fp32 layout: lane&15 = N, VGPR r -> M = r + 8*(lane>>4)
        const int coln = j * 16 + l16;
        #pragma unroll
        for (int r = 0; r < 8; ++r) {
            const int m   = r + 8 * half;
            const int row = i * 16 + m;
            float v = c[r] * scale;
            if (coln > row) v = -INFINITY;   // mask inside diagonal tile
            sc[m][coln] = v;
        }
    }
    __syncthreads();

    // ---- in-place softmax: 8 lanes per row, shuffle reductions ----
    {
        const int m     = t >> 3;            // row within tile (0..15)
        const int sub   = t & 7;
        const int ncols = i * 16 + 16;       // valid prefix incl. masked -inf
        float mx = -INFINITY;
        for (int c = sub; c < ncols; c += 8) mx = fmaxf(mx, sc[m][c]);
        #pragma unroll
        for (int off = 4; off >= 1; off >>= 1)
            mx = fmaxf(mx, __shfl_xor(mx, off, 8));
        float sum = 0.f;
        for (int c = sub; c < ncols; c += 8) {
            float e = __expf(sc[m][c] - mx);
            sc[m][c] = e;
            sum += e;
        }
        #pragma unroll
        for (int off = 4; off >= 1; off >>= 1)
            sum += __shfl_xor(sum, off, 8);
        const float inv = 1.f / sum;
        for (int c = sub; c < ncols; c += 8) sc[m][c] *= inv;
    }
    __syncthreads();

    // ---- dropout + single streaming write (bandwidth-bound phase) ----
    float* orow = out + ((size_t)(b * TDIM) + i * 16) * TDIM;
    for (int e = t; e < 16 * TDIM; e += BLOCK) {
        const int m   = e >> 10;             // TDIM == 1024
        const int col = e & (TDIM - 1);
        const int row = i * 16 + m;
        float p = 0.f;
        if (col <= row) {
            p = sc[m][col];
            // deterministic dropout mask: splitmix64(seed=42, linear index)
            uint64_t x = (((uint64_t)(b * TDIM + row)) << 32) | (uint32_t)col;
            x += 0x9E3779B97F4A7C15ull * 43ull;           // seed 42 offset
            x = (x ^ (x >> 30)) * 0xBF58476D1CE4E5B9ull;
            x = (x ^ (x >> 27)) * 0x94D049BB133111EBull;
            x ^= (x >> 31);
            const float u = (float)(uint32_t)((x >> 40) & 0xFFFFFFull)
                            * (1.0f / 16777216.0f);
            p = (u < 0.8f) ? p * 1.25f : 0.f;  // keep-prob 0.8, scale 1/0.8
        }
        orow[(size_t)m * TDIM + col] = p;
    }
}

extern "C" void kernel_launch(void* const* d_in, const int* in_sizes, int n_in,
                              void* d_out, int out_size, void* d_ws, size_t ws_size,
                              hipStream_t stream) {
    (void)n_in; (void)out_size; (void)d_ws; (void)ws_size;
    const float* q = (const float*)d_in[0];
    const float* k = (const float*)d_in[1];
    float* out = (float*)d_out;
    const int B = in_sizes[0] / (TDIM * DDIM);   // 4 for the reference shapes
    dim3 grid(TDIM / 16, B);
    attn_probs_dropout_kernel<<<grid, BLOCK, 0, stream>>>(q, k, out);
}